// LIF_R_ASC_no_grad_10316511445824
// MI455X (gfx1250) — compile-verified
//
#include <hip/hip_runtime.h>

// LIF-R-ASC recurrence on MI455X (gfx1250, wave32).
// One wave owns a 16-batch tile; all state lives in VGPRs across T=1024 steps.
// Recurrent matmul: I^T = Wm^T @ act^T via 3x chained V_WMMA_F32_16X16X4_F32.
// x / soft are pure 100MB streams -> non-temporal loads/stores.

typedef float v2f __attribute__((ext_vector_type(2)));
typedef float v4f __attribute__((ext_vector_type(4)));
typedef float v8f __attribute__((ext_vector_type(8)));

#define T_STEPS 1024
#define BATCH   2048
#define NN      12
#define ROW_STRIDE (BATCH * NN)   // floats per timestep in x / out

// lane <-> lane^16 exchange (group-of-32 swizzle: and=0x1F, or=0, xor=0x10)
__device__ __forceinline__ float swap16(float x) {
  return __int_as_float(__builtin_amdgcn_ds_swizzle(__float_as_int(x), 0x401F));
}

// Load this lane's 8 neuron inputs for one (t, batch) row (streaming, NT hint).
// h==0 lanes: neurons 0..7 (two b128), h==1 lanes: neurons 8..11 (one b128).
__device__ __forceinline__ void loadx(float xr[8], const float* __restrict__ row, int h) {
  const v4f* p = (const v4f*)(row + 8 * h);
  v4f a = __builtin_nontemporal_load(p);
  xr[0] = a.x; xr[1] = a.y; xr[2] = a.z; xr[3] = a.w;
  if (h == 0) {
    v4f b = __builtin_nontemporal_load(p + 1);
    xr[4] = b.x; xr[5] = b.y; xr[6] = b.z; xr[7] = b.w;
  } else {
    xr[4] = 0.f; xr[5] = 0.f; xr[6] = 0.f; xr[7] = 0.f;
  }
}

__global__ __launch_bounds__(128) void lif_wmma_kernel(
    const float* __restrict__ x,      const float* __restrict__ w,
    const float* __restrict__ E_L,    const float* __restrict__ tau_m,
    const float* __restrict__ tau_s,  const float* __restrict__ Gp,
    const float* __restrict__ f_v,    const float* __restrict__ f_I,
    const float* __restrict__ dths,   const float* __restrict__ b_s,
    const float* __restrict__ dV,     const float* __restrict__ normR,
    const int*   __restrict__ ntypes, float* __restrict__ out)
{
  const int lane = threadIdx.x & 31;
  const int wave = (blockIdx.x * blockDim.x + threadIdx.x) >> 5;  // 0..127 tile id
  const int h    = lane >> 4;        // lane half: neuron block 0..7 vs 8..15
  const int col  = lane & 15;        // batch within tile == B/D column == A row
  const long long bglob = (long long)wave * 16 + col;

  // ---- A = Wm^T, 3 K=4 slices, loop-invariant ----
  // A-layout (16x4 f32): lanes 0-15 & 16-31 both M=lane%16; VGPR j holds K = j + 2*(lane/16).
  // Wm^T[m,k] = w[k,m] * sign(neuron_type[k]), zero diagonal, zero pad rows m>=12.
  v2f A[3];
#pragma unroll
  for (int kb = 0; kb < 3; ++kb) {
#pragma unroll
    for (int j = 0; j < 2; ++j) {
      int k = 4 * kb + j + 2 * h;        // input-neuron (K dim), always <= 11
      float val = 0.f;
      if (col < NN && k != col)
        val = w[k * NN + col] * (float)ntypes[k];
      A[kb][j] = val;
    }
  }

  // ---- per-lane neuron constants + state (neuron = 8*h + r) ----
  float cEL[8], cITm[8], cITs[8], cG[8], cFv[8], cFi[8], cDth[8], cBs[8], cDV[8], cNm[8];
  float v[8], s[8], th[8], Ia[8];
#pragma unroll
  for (int r = 0; r < 8; ++r) {
    int  nn = 8 * h + r;
    bool ok = nn < NN;
    int  ni = ok ? nn : 0;
    cEL[r]  = ok ? E_L[ni]        : 0.f;
    cITm[r] = ok ? 1.f / tau_m[ni] : 1.f;
    cITs[r] = ok ? 1.f / tau_s[ni] : 1.f;
    cG[r]   = ok ? Gp[ni]   : 0.f;
    cFv[r]  = ok ? f_v[ni]  : 0.f;
    cFi[r]  = ok ? f_I[ni]  : 0.f;
    cDth[r] = ok ? dths[ni] : 0.f;
    cBs[r]  = ok ? b_s[ni]  : 0.f;
    cDV[r]  = ok ? dV[ni]   : 0.f;
    cNm[r]  = ok ? normR[ni] : 0.f;
    v[r]  = cEL[r];
    s[r]  = 0.f;
    th[r] = ok ? dths[ni] : 1.f;   // keep padded theta nonzero (rcp-safe)
    Ia[r] = 0.f;
  }

  // Bumped streaming pointers (avoid per-step t*stride multiplies).
  const float* xq = x   + bglob * NN + ROW_STRIDE;  // next-step prefetch source
  float*       oq = out + bglob * NN;               // this-step store target

  float xr[8];
  loadx(xr, x + bglob * NN, h);                     // t = 0

  for (int t = 0; t < T_STEPS; ++t) {
    // Accumulator pre-seeded with the feed-forward current 1.75 * x_t
    v8f C;
#pragma unroll
    for (int r = 0; r < 8; ++r) C[r] = 1.75f * xr[r];

    // Prefetch next timestep's x behind the WMMA + neuron-update chain.
    if (t + 1 < T_STEPS) loadx(xr, xq, h);
    xq += ROW_STRIDE;

    // act = 0.5 * (Ia + s)
    float act[8];
#pragma unroll
    for (int r = 0; r < 8; ++r) act[r] = 0.5f * (Ia[r] + s[r]);

    // B-layout (4x16 f32): VGPR j, lane half hh holds row k' = j + 2*hh, col = lane%16 (batch).
    // Lane owns neurons 8h..8h+7; off-half rows come from lane^16 via ds_swizzle.
    {  // K slice 0: k = j + 2h   (h=0 local act[j]; h=1 remote act[2+j])
      float s2 = swap16(act[2]), s3 = swap16(act[3]);
      v2f B; B[0] = h ? s2 : act[0]; B[1] = h ? s3 : act[1];
      C = __builtin_amdgcn_wmma_f32_16x16x4_f32(false, A[0], false, B, (short)0, C, false, false);
    }
    {  // K slice 1: k = 4 + j + 2h   (h=0 local act[4+j]; h=1 remote act[6+j])
      float s6 = swap16(act[6]), s7 = swap16(act[7]);
      v2f B; B[0] = h ? s6 : act[4]; B[1] = h ? s7 : act[5];
      C = __builtin_amdgcn_wmma_f32_16x16x4_f32(false, A[1], false, B, (short)0, C, false, false);
    }
    {  // K slice 2: k = 8 + j + 2h   (h=0 remote act[j]; h=1 local act[2+j])
      float s0 = swap16(act[0]), s1 = swap16(act[1]);
      v2f B; B[0] = h ? act[2] : s0; B[1] = h ? act[3] : s1;
      C = __builtin_amdgcn_wmma_f32_16x16x4_f32(false, A[2], false, B, (short)0, C, false, false);
    }

    // ---- neuron dynamics (8 independent (batch,neuron) elements per lane) ----
    float soft[8];
#pragma unroll
    for (int r = 0; r < 8; ++r) {
      float I   = C[r];
      float dv  = (cG[r] * (cEL[r] - v[r]) + I * cNm[r]) * cITm[r];
      float vn  = v[r] + dv;
      bool  sp  = vn >= th[r];
      float gating = fminf(fmaxf(vn * __builtin_amdgcn_rcpf(th[r]), 0.f), 1.f);
      float dscl   = fminf(fmaxf(dv * __builtin_amdgcn_rcpf(th[r] - cEL[r]), 0.f), 1.f);
      s[r] += (-s[r] + gating * dscl) * cITs[r];
      float vreset = cEL[r] + cFv[r] * (v[r] - cEL[r]) - cDV[r];
      v[r]  = sp ? vreset : vn;
      float thn = sp ? (th[r] + cDth[r]) : (1.f - cBs[r]) * th[r];
      th[r] = thn;
      Ia[r] = Ia[r] * (1.f - cFi[r]) + (sp ? cFi[r] : 0.f);
      // sigmoid(vn - thn), using the UPDATED theta per the reference
      soft[r] = __builtin_amdgcn_rcpf(1.f + __expf(thn - vn));
    }

    // Streaming stores with NT hint (written once, never re-read).
    v4f o0; o0.x = soft[0]; o0.y = soft[1]; o0.z = soft[2]; o0.w = soft[3];
    __builtin_nontemporal_store(o0, (v4f*)(oq + 8 * h));
    if (h == 0) {
      v4f o1; o1.x = soft[4]; o1.y = soft[5]; o1.z = soft[6]; o1.w = soft[7];
      __builtin_nontemporal_store(o1, (v4f*)(oq + 4));
    }
    oq += ROW_STRIDE;
  }
}

extern "C" void kernel_launch(void* const* d_in, const int* in_sizes, int n_in,
                              void* d_out, int out_size, void* d_ws, size_t ws_size,
                              hipStream_t stream) {
  const float* x      = (const float*)d_in[0];
  const float* w      = (const float*)d_in[1];
  const float* E_L    = (const float*)d_in[2];
  const float* tau_m  = (const float*)d_in[3];
  const float* tau_s  = (const float*)d_in[4];
  const float* G      = (const float*)d_in[5];
  const float* f_v    = (const float*)d_in[6];
  const float* f_I    = (const float*)d_in[7];
  const float* dths   = (const float*)d_in[8];
  const float* b_s    = (const float*)d_in[9];
  const float* dV     = (const float*)d_in[10];
  const float* normR  = (const float*)d_in[11];
  const int*   ntypes = (const int*)d_in[12];
  float*       out    = (float*)d_out;

  // 2048 batch / 16 per wave = 128 waves; 4 waves (128 threads) per block.
  dim3 grid(32), block(128);
  hipLaunchKernelGGL(lif_wmma_kernel, grid, block, 0, stream,
                     x, w, E_L, tau_m, tau_s, G, f_v, f_I, dths, b_s, dV, normR,
                     ntypes, out);
}